// NGPSpacedSampler_47175920779939
// MI455X (gfx1250) — compile-verified
//
#include <hip/hip_runtime.h>
#include <hip/hip_bf16.h>
#include <stdint.h>

// ---------------------------------------------------------------------------
// MI455X reasoning: bandwidth-bound compaction (~0.3-0.4 GB moved, ~250 MFLOP)
// => ~15-20us at 23.3 TB/s. No matrix ops -> no WMMA. Optimize with wave32
// ballot compaction, b128 vector memory, tail-only zeroing, and the gfx1250
// async global->LDS path (GLOBAL_LOAD_ASYNC_TO_LDS_B32 + s_wait_asynccnt).
// ---------------------------------------------------------------------------

#define T_CLAMP 1e10f
#define WPB 8  // waves (rays) per 256-thread block

#if __has_builtin(__builtin_amdgcn_global_load_async_to_lds_b32) && \
    __has_builtin(__builtin_amdgcn_s_wait_asynccnt)
#define HAVE_ASYNC_LDS 1
#else
#define HAVE_ASYNC_LDS 0
#endif

typedef __attribute__((address_space(1))) int* gint_ptr;  // clang prints as '__device__ int *'
typedef __attribute__((address_space(3))) int* lint_ptr;  // clang prints as '__shared__ int *'

__device__ __forceinline__ unsigned wave_ballot32(bool p) {
#if __has_builtin(__builtin_amdgcn_ballot_w32)
  return __builtin_amdgcn_ballot_w32(p);
#else
  return (unsigned)__ballot(p);  // wave32: low 32 bits
#endif
}

#if HAVE_ASYNC_LDS
__device__ __forceinline__ void async_g2l_b32(const float* g, float* l) {
  // gfx1250 GLOBAL_LOAD_ASYNC_TO_LDS_B32 (tracked by ASYNCcnt)
  __builtin_amdgcn_global_load_async_to_lds_b32(
      (gint_ptr)(void*)g, (lint_ptr)(void*)l, 0, 0);
}
#endif

// ---------------------------------------------------------------------------
// Kernel A: one wave per ray. Slab test, 128 occupancy gathers, ballot masks.
// ---------------------------------------------------------------------------
__global__ void k_march_count(const float* __restrict__ ro,
                              const float* __restrict__ rd,
                              const float* __restrict__ aabb,
                              const float* __restrict__ grid,
                              float* __restrict__ tmin_out,
                              float* __restrict__ tmax_out,
                              uint4* __restrict__ masks,
                              int R, int S, int res) {
  __shared__ float s_o[WPB * 3];
  __shared__ float s_d[WPB * 3];

  const int tid = threadIdx.x;
  const int rayBase = blockIdx.x * WPB;
  const int nflt = min(WPB, R - rayBase) * 3;

  // Stage this block's ray origins/dirs into LDS.
#if HAVE_ASYNC_LDS
  if (tid < 24 && tid < nflt)
    async_g2l_b32(ro + (size_t)rayBase * 3 + tid, &s_o[tid]);
  if (tid >= 64 && tid < 88 && (tid - 64) < nflt)
    async_g2l_b32(rd + (size_t)rayBase * 3 + (tid - 64), &s_d[tid - 64]);
  __builtin_amdgcn_s_wait_asynccnt(0);
#else
  if (tid < 24 && tid < nflt) s_o[tid] = ro[(size_t)rayBase * 3 + tid];
  if (tid >= 64 && tid < 88 && (tid - 64) < nflt)
    s_d[tid - 64] = rd[(size_t)rayBase * 3 + (tid - 64)];
#endif
  // Speculative prefetch of next block's ray data (global_prefetch_b8).
  if (tid == 0 && rayBase + WPB < R) {
    __builtin_prefetch(ro + (size_t)(rayBase + WPB) * 3, 0, 1);
    __builtin_prefetch(rd + (size_t)(rayBase + WPB) * 3, 0, 1);
  }
  __syncthreads();

  const int wave = tid >> 5;
  const int lane = tid & 31;
  const int ray = rayBase + wave;
  if (ray >= R) return;

  const float ox = s_o[wave * 3 + 0], oy = s_o[wave * 3 + 1], oz = s_o[wave * 3 + 2];
  const float dx = s_d[wave * 3 + 0], dy = s_d[wave * 3 + 1], dz = s_d[wave * 3 + 2];
  const float a0 = aabb[0], a1 = aabb[1], a2 = aabb[2];
  const float b0 = aabb[3], b1 = aabb[4], b2 = aabb[5];

  const float ix = 1.0f / dx, iy = 1.0f / dy, iz = 1.0f / dz;
  const float t0x = (a0 - ox) * ix, t1x = (b0 - ox) * ix;
  const float t0y = (a1 - oy) * iy, t1y = (b1 - oy) * iy;
  const float t0z = (a2 - oz) * iz, t1z = (b2 - oz) * iz;
  float tn = fmaxf(fmaxf(fminf(t0x, t1x), fminf(t0y, t1y)), fminf(t0z, t1z));
  float tf = fminf(fminf(fmaxf(t0x, t1x), fmaxf(t0y, t1y)), fmaxf(t0z, t1z));
  tn = fmaxf(tn, 0.0f);
  const bool miss = (tf <= tn);
  const float tmin = fminf(miss ? T_CLAMP : tn, T_CLAMP);
  const float tmax = fminf(miss ? T_CLAMP : tf, T_CLAMP);
  const float dt = miss ? 0.0f : (tmax - tmin) / (float)S;
  const bool live = (!miss) && (dt > 0.0f);

  const float resf = (float)res;
  const float sx = resf / (b0 - a0), sy = resf / (b1 - a1), sz = resf / (b2 - a2);
  const int rmax = res - 1;

  unsigned m[4];
#pragma unroll
  for (int j = 0; j < 4; ++j) {
    bool v = false;
    const int s = j * 32 + lane;
    if (live && s < S) {
      const float st = tmin + (float)s * dt;
      const float en = st + dt;
      const float tm = 0.5f * (st + en);
      const float px = ox + dx * tm, py = oy + dy * tm, pz = oz + dz * tm;
      int cx = (int)floorf((px - a0) * sx);
      int cy = (int)floorf((py - a1) * sy);
      int cz = (int)floorf((pz - a2) * sz);
      cx = cx < 0 ? 0 : (cx > rmax ? rmax : cx);
      cy = cy < 0 ? 0 : (cy > rmax ? rmax : cy);
      cz = cz < 0 ? 0 : (cz > rmax ? rmax : cz);
      const float g = grid[((size_t)cx * res + cy) * res + cz];
      v = g > 0.5f;
    }
    m[j] = wave_ballot32(v);
  }

  if (lane == 0) {
    tmin_out[ray] = tmin;
    tmax_out[ray] = tmax;
    masks[ray] = make_uint4(m[0], m[1], m[2], m[3]);
  }
}

// ---------------------------------------------------------------------------
// Kernel B: single-block exclusive scan of per-ray popcounts -> ray_start.
// ---------------------------------------------------------------------------
__global__ void k_scan(const uint4* __restrict__ masks,
                       int* __restrict__ ray_start,
                       int* __restrict__ nvalid, int R) {
  __shared__ int sh[1024];
  const int t = threadIdx.x;
  const int chunk = (R + 1023) >> 10;
  const int begin = t * chunk;
  const int end = min(begin + chunk, R);

  int local = 0;
  for (int r = begin; r < end; ++r) {
    const uint4 m = masks[r];
    local += __popc(m.x) + __popc(m.y) + __popc(m.z) + __popc(m.w);
  }
  sh[t] = local;
  __syncthreads();

  for (int off = 1; off < 1024; off <<= 1) {
    const int add = (t >= off) ? sh[t - off] : 0;
    __syncthreads();
    sh[t] += add;
    __syncthreads();
  }

  int run = (t == 0) ? 0 : sh[t - 1];
  for (int r = begin; r < end; ++r) {
    const uint4 m = masks[r];
    ray_start[r] = run;
    run += __popc(m.x) + __popc(m.y) + __popc(m.z) + __popc(m.w);
  }
  if (t == 1023) nvalid[0] = sh[1023];
}

// ---------------------------------------------------------------------------
// Kernel C: scatter valid samples (two b128 stores each) + packed_info.
// ---------------------------------------------------------------------------
__global__ void k_scatter(const float* __restrict__ ro,
                          const float* __restrict__ rd,
                          const float* __restrict__ tmin_arr,
                          const float* __restrict__ tmax_arr,
                          const uint4* __restrict__ masks,
                          const int* __restrict__ ray_start,
                          float* __restrict__ packed,
                          float* __restrict__ info,
                          int R, int S) {
  const int tid = threadIdx.x;
  const int wave = tid >> 5;
  const int lane = tid & 31;
  const int ray = blockIdx.x * WPB + wave;
  if (ray >= R) return;

  const uint4 m = masks[ray];
  const int base = ray_start[ray];
  const int c0 = __popc(m.x), c1 = __popc(m.y), c2 = __popc(m.z), c3 = __popc(m.w);

  const float tmin = tmin_arr[ray], tmax = tmax_arr[ray];
  const float dt = (tmax - tmin) / (float)S;  // 0 for miss rays (masks are 0)
  const float ox = ro[(size_t)ray * 3 + 0], oy = ro[(size_t)ray * 3 + 1], oz = ro[(size_t)ray * 3 + 2];
  const float dx = rd[(size_t)ray * 3 + 0], dy = rd[(size_t)ray * 3 + 1], dz = rd[(size_t)ray * 3 + 2];

  const unsigned below = (1u << lane) - 1u;
  const unsigned mm[4] = {m.x, m.y, m.z, m.w};
  const int pre[4] = {base, base + c0, base + c0 + c1, base + c0 + c1 + c2};

#pragma unroll
  for (int j = 0; j < 4; ++j) {
    const int s = j * 32 + lane;
    if (s < S && ((mm[j] >> lane) & 1u)) {
      const int rank = pre[j] + __popc(mm[j] & below);
      const float st = tmin + (float)s * dt;
      const float en = st + dt;
      float* dst = packed + (size_t)rank * 8;
      *reinterpret_cast<float4*>(dst) = make_float4(ox, oy, oz, dx);
      *reinterpret_cast<float4*>(dst + 4) = make_float4(dy, dz, st, en);
    }
  }

  if (lane == 0) {
    info[(size_t)ray * 2 + 0] = (float)base;              // exact: < 2^24
    info[(size_t)ray * 2 + 1] = (float)(c0 + c1 + c2 + c3);
  }
}

// ---------------------------------------------------------------------------
// Kernel D: zero only the tail (slots >= nvalid) of the packed buffer.
// ---------------------------------------------------------------------------
__global__ void k_zero_tail(float* __restrict__ packed,
                            const int* __restrict__ nvalid, int total) {
  const int slot = blockIdx.x * blockDim.x + threadIdx.x;
  if (slot >= total) return;
  if (slot >= nvalid[0]) {
    float4* p = reinterpret_cast<float4*>(packed + (size_t)slot * 8);
    const float4 z = make_float4(0.f, 0.f, 0.f, 0.f);
    p[0] = z;
    p[1] = z;
  }
}

// ---------------------------------------------------------------------------
extern "C" void kernel_launch(void* const* d_in, const int* in_sizes, int n_in,
                              void* d_out, int out_size, void* d_ws, size_t ws_size,
                              hipStream_t stream) {
  const float* ro = (const float*)d_in[0];
  const float* rd = (const float*)d_in[1];
  const float* aabb = (const float*)d_in[2];
  const float* grid = (const float*)d_in[3];

  const int R = in_sizes[0] / 3;
  const int gn = in_sizes[3];
  int res = 1;
  while ((long long)res * res * res < (long long)gn) ++res;  // 128^3 -> 128
  // out_size = R*S*8 + 2R + R + R = R*(8S+4)  =>  recover S on host.
  const int S = (int)(((long long)out_size / R - 4) / 8);
  const int total = R * S;

  float* packed = (float*)d_out;                       // [total, 8]
  float* info = packed + (size_t)total * 8;            // [R, 2] (as f32)
  float* tmin = info + (size_t)R * 2;                  // [R]
  float* tmax = tmin + R;                              // [R]

  uint4* masks = (uint4*)d_ws;                                   // R * 16 B
  int* ray_start = (int*)((char*)d_ws + (size_t)R * 16);         // R * 4 B
  int* nvalid = ray_start + R;                                   // 4 B

  const int blocksA = (R + WPB - 1) / WPB;
  hipLaunchKernelGGL(k_march_count, dim3(blocksA), dim3(256), 0, stream,
                     ro, rd, aabb, grid, tmin, tmax, masks, R, S, res);
  hipLaunchKernelGGL(k_scan, dim3(1), dim3(1024), 0, stream,
                     masks, ray_start, nvalid, R);
  hipLaunchKernelGGL(k_scatter, dim3(blocksA), dim3(256), 0, stream,
                     ro, rd, tmin, tmax, masks, ray_start, packed, info, R, S);
  const int blocksD = (total + 255) / 256;
  hipLaunchKernelGGL(k_zero_tail, dim3(blocksD), dim3(256), 0, stream,
                     packed, nvalid, total);
}